// FermiNet_60790967107807
// MI455X (gfx1250) — compile-verified
//
#include <hip/hip_runtime.h>
#include <math.h>

typedef __attribute__((ext_vector_type(2))) float v2f;
typedef __attribute__((ext_vector_type(8))) float v8f;

#define NELEC 16
#define IN1   32
#define IN2   8
#define FLEN  112
#define TB    16          // WMMA M-tile (batches)
#define MTA   4           // M-tiles per block, double kernel  -> 64 batches/block
#define MTB   2           // M-tiles per block, single kernel  -> 32 batches/block
#define DH_B  2048        // 16*16*8 floats per batch in double_h
#define SH_B  512         // 16*32 floats per batch in single_h

// Branch-free tanh: sign(x) * (1-u)/(1+u), u = exp(-2|x|).
// 1-u is Sterbenz-exact -> abs error ~3e-8; saturates to +-1; +-0/NaN preserved.
__device__ __forceinline__ float fast_tanh(float x) {
    float ax = __builtin_fabsf(x);
    float u  = __expf(-2.0f * ax);                       // v_mul + v_exp_f32
    float t  = (1.0f - u) * __builtin_amdgcn_rcpf(1.0f + u);
    return __builtin_copysignf(t, x);
}

// ---------------------------------------------------------------------------
// Kernel A: g2 means (optional, to workspace) + double stream
//   out[b,n,m,i] = tanh( sum_j W[n,m,i,j] * h[b,n,m,j] + c[n,m,i] ) + h[b,n,m,i]
// Two adjacent pairs (m0, m0+1) packed into ONE dense 16x16x16 WMMA problem:
//   A: K=0..7 <- h[.,n,m0,:], K=8..15 <- h[.,n,m0+1,:]   (16 contiguous floats)
//   B: block-diagonal [ W[n,m0]^T  0 ; 0  W[n,m0+1]^T ]
// -> every lane/column is useful: dense epilogue, no exec divergence.
// B fragments reused across MTA batch tiles.
// ---------------------------------------------------------------------------
__global__ __launch_bounds__(256) void fermi_double_kernel(
    const float* __restrict__ dh,    // (B,16,16,8)
    const float* __restrict__ w,     // (16,16,8,8)
    const float* __restrict__ c,     // (16,16,8)
    float* __restrict__ out_double,  // (B,16,16,8)
    float* __restrict__ g2u, float* __restrict__ g2d,   // (B,16,8) each
    int write_g2)
{
    const int b0  = blockIdx.x * (TB * MTA);
    const int tid = threadIdx.x;
    const float* dhb = dh + (size_t)b0 * DH_B;

    if (write_g2) {
        for (int idx = tid; idx < TB * MTA * 128; idx += 256) {
            int bl = idx >> 7, mk = idx & 127;
            const float* p = dhb + (size_t)bl * DH_B + mk;
            float su = 0.f, sd = 0.f;
            #pragma unroll
            for (int n = 0; n < 8; ++n)  su += p[n * 128];
            #pragma unroll
            for (int n = 8; n < 16; ++n) sd += p[n * 128];
            g2u[(size_t)(b0 + bl) * 128 + mk] = su * 0.125f;
            g2d[(size_t)(b0 + bl) * 128 + mk] = sd * 0.125f;
        }
    }

    const int wave  = tid >> 5;
    const int lane  = tid & 31;
    const int lhalf = lane >> 4;
    const int l15   = lane & 15;
    const int icol  = l15;                    // N column 0..15 (m0: 0-7, m1: 8-15)
    const int ic    = icol & 7;               // feature index within its pair
    const float mask0 = (icol < 8) ? 1.0f : 0.0f;   // column belongs to m0 block
    const float mask1 = 1.0f - mask0;               // column belongs to m1 block

    // 128 m-pairs total, 16 per wave
    for (int pp = 0; pp < 16; ++pp) {
        const int p  = wave * 16 + pp;
        const int n  = p >> 3;
        const int m0 = (p & 7) * 2;
        const float* wnm0 = w + (size_t)(n * 16 + m0) * 64;   // W[n][m0..m0+1]: 128 floats

        // Block-diagonal B fragments: loaded once per m-pair, reused over MTA tiles.
        v2f bm[4];
        #pragma unroll
        for (int kc = 0; kc < 4; ++kc) {
            const int ks = kc * 4 + 2 * lhalf;                // 0..15, never straddles 8
            const float msk = (ks < 8) ? mask0 : mask1;       // compile-time select
            const int base  = (ks < 8) ? (ic * 8 + ks)
                                       : (64 + ic * 8 + (ks - 8));
            bm[kc].x = wnm0[base]     * msk;
            bm[kc].y = wnm0[base + 1] * msk;
        }

        v8f acc[MTA];
        #pragma unroll
        for (int mt = 0; mt < MTA; ++mt) {
            acc[mt] = (v8f){};
            // A rows: 16 contiguous floats = h[b,n,m0,0:8] ++ h[b,n,m0+1,0:8]
            const float* hrow = dhb + (size_t)(mt * 16 + l15) * DH_B + n * 128 + m0 * 8;
            #pragma unroll
            for (int kc = 0; kc < 4; ++kc) {
                const int ks = kc * 4 + 2 * lhalf;
                v2f a;  a.x = hrow[ks];  a.y = hrow[ks + 1];
                acc[mt] = __builtin_amdgcn_wmma_f32_16x16x4_f32(
                              false, a, false, bm[kc], (short)0, acc[mt], false, false);
            }
        }

        // c for columns 0..15 of this m-pair is contiguous
        const float cv = c[(size_t)(n * 16 + m0) * 8 + icol];
        #pragma unroll
        for (int mt = 0; mt < MTA; ++mt) {
            #pragma unroll
            for (int v = 0; v < 8; ++v) {
                const int M = mt * 16 + v + 8 * lhalf;        // batch row in block
                const size_t off = (size_t)(b0 + M) * DH_B + n * 128 + m0 * 8 + icol;
                out_double[off] = fast_tanh(acc[mt][v] + cv) + dh[off];
            }
        }
    }
}

// ---------------------------------------------------------------------------
// Kernel B: g1 means (LDS) + g2 (ws or recompute) + single stream
//   out[b,n,o] = tanh( sum_f f[b,n,f] * V[n,f,o] + bvec[n,o] ) + sh[b,n,o]
// WMMA: M = 16 batches (x MTB tiles, V fragment reused), K = 112, N = 32 (2 tiles)
// ---------------------------------------------------------------------------
__global__ __launch_bounds__(256) void fermi_single_kernel(
    const float* __restrict__ sh,     // (B,16,32)
    const float* __restrict__ dh,     // (B,16,16,8)  (fallback g2 source)
    const float* __restrict__ vmat,   // (16,112,32)
    const float* __restrict__ bvec,   // (16,32)
    const float* __restrict__ g2u_ws, const float* __restrict__ g2d_ws,
    float* __restrict__ out_single,   // (B,16,32)
    int g2_from_ws)
{
    __shared__ float g1u[TB * MTB * 36];    // stride 36: conflict-free row reads
    __shared__ float g1d[TB * MTB * 36];
    __shared__ float g2u[TB * MTB * 132];   // stride 132: conflict-free
    __shared__ float g2d[TB * MTB * 132];

    const int b0  = blockIdx.x * (TB * MTB);
    const int tid = threadIdx.x;
    const float* shb = sh + (size_t)b0 * SH_B;

    // spin-wise g1 means over electrons
    for (int idx = tid; idx < TB * MTB * IN1; idx += 256) {
        int bl = idx >> 5, d = idx & 31;
        const float* p = shb + (size_t)bl * SH_B + d;
        float su = 0.f, sd = 0.f;
        #pragma unroll
        for (int n = 0; n < 8; ++n)  su += p[n * 32];
        #pragma unroll
        for (int n = 8; n < 16; ++n) sd += p[n * 32];
        g1u[bl * 36 + d] = su * 0.125f;
        g1d[bl * 36 + d] = sd * 0.125f;
    }
    // g2 means: from workspace, or recompute from double_h (L2-resident)
    if (g2_from_ws) {
        for (int idx = tid; idx < TB * MTB * 128; idx += 256) {
            int bl = idx >> 7, mk = idx & 127;
            g2u[bl * 132 + mk] = g2u_ws[(size_t)(b0 + bl) * 128 + mk];
            g2d[bl * 132 + mk] = g2d_ws[(size_t)(b0 + bl) * 128 + mk];
        }
    } else {
        for (int idx = tid; idx < TB * MTB * 128; idx += 256) {
            int bl = idx >> 7, mk = idx & 127;
            const float* p = dh + (size_t)(b0 + bl) * DH_B + mk;
            float su = 0.f, sd = 0.f;
            #pragma unroll
            for (int n = 0; n < 8; ++n)  su += p[n * 128];
            #pragma unroll
            for (int n = 8; n < 16; ++n) sd += p[n * 128];
            g2u[bl * 132 + mk] = su * 0.125f;
            g2d[bl * 132 + mk] = sd * 0.125f;
        }
    }
    __syncthreads();

    const int wave  = tid >> 5;
    const int lane  = tid & 31;
    const int lhalf = lane >> 4;
    const int l15   = lane & 15;

    for (int ni = 0; ni < 2; ++ni) {
        const int n = wave * 2 + ni;
        const float* vn = vmat + (size_t)n * FLEN * 32;

        // per-M-tile f-segment base pointers
        const float* srow[MTB];
        const float* g1urow[MTB];
        const float* g1drow[MTB];
        const float* g2urow[MTB];
        const float* g2drow[MTB];
        #pragma unroll
        for (int mt = 0; mt < MTB; ++mt) {
            const int row = mt * 16 + l15;
            srow[mt]   = shb + (size_t)row * SH_B + n * 32;   // f[0:32]
            g1urow[mt] = &g1u[row * 36];                      // f[32:64]
            g1drow[mt] = &g1d[row * 36];                      // f[64:96]
            g2urow[mt] = &g2u[row * 132 + n * 8];             // f[96:104]
            g2drow[mt] = &g2d[row * 132 + n * 8];             // f[104:112]
        }

        v8f acc[MTB][2];
        #pragma unroll
        for (int mt = 0; mt < MTB; ++mt) { acc[mt][0] = (v8f){}; acc[mt][1] = (v8f){}; }

        #pragma unroll
        for (int kk = 0; kk < 28; ++kk) {
            const int k0 = kk * 4;
            const int ks = k0 + 2 * lhalf;
            // B fragments (V): loaded ONCE, reused across MTB batch tiles
            const float* vk = vn + (size_t)ks * 32;           // B[k][o] = V[n,k,o]
            v2f bm0, bm1;
            bm0.x = vk[l15];       bm0.y = vk[32 + l15];      // o-tile 0
            bm1.x = vk[16 + l15];  bm1.y = vk[48 + l15];      // o-tile 1
            #pragma unroll
            for (int mt = 0; mt < MTB; ++mt) {
                v2f a;
                if (k0 < 32)       { a.x = srow[mt][ks];         a.y = srow[mt][ks + 1]; }
                else if (k0 < 64)  { a.x = g1urow[mt][ks - 32];  a.y = g1urow[mt][ks - 31]; }
                else if (k0 < 96)  { a.x = g1drow[mt][ks - 64];  a.y = g1drow[mt][ks - 63]; }
                else if (k0 < 104) { a.x = g2urow[mt][ks - 96];  a.y = g2urow[mt][ks - 95]; }
                else               { a.x = g2drow[mt][ks - 104]; a.y = g2drow[mt][ks - 103]; }
                acc[mt][0] = __builtin_amdgcn_wmma_f32_16x16x4_f32(
                                 false, a, false, bm0, (short)0, acc[mt][0], false, false);
                acc[mt][1] = __builtin_amdgcn_wmma_f32_16x16x4_f32(
                                 false, a, false, bm1, (short)0, acc[mt][1], false, false);
            }
        }

        const float bb0 = bvec[n * 32 + l15];
        const float bb1 = bvec[n * 32 + 16 + l15];
        #pragma unroll
        for (int mt = 0; mt < MTB; ++mt) {
            #pragma unroll
            for (int v = 0; v < 8; ++v) {
                const int M = mt * 16 + v + 8 * lhalf;
                const size_t base = (size_t)(b0 + M) * SH_B + n * 32;
                out_single[base + l15]      = fast_tanh(acc[mt][0][v] + bb0) + sh[base + l15];
                out_single[base + 16 + l15] = fast_tanh(acc[mt][1][v] + bb1) + sh[base + 16 + l15];
            }
        }
    }
}

// ---------------------------------------------------------------------------
extern "C" void kernel_launch(void* const* d_in, const int* in_sizes, int n_in,
                              void* d_out, int out_size, void* d_ws, size_t ws_size,
                              hipStream_t stream) {
    const float* sh = (const float*)d_in[0];   // single_h (B,16,32)
    const float* dh = (const float*)d_in[1];   // double_h (B,16,16,8)
    const float* vm = (const float*)d_in[2];   // v_matrices (1,16,112,32)
    const float* bv = (const float*)d_in[3];   // b_vectors (1,16,32)
    const float* wm = (const float*)d_in[4];   // w_matrices (1,16,16,8,8)
    const float* cv = (const float*)d_in[5];   // c_vectors (1,16,16,8)

    const int B = in_sizes[0] / (NELEC * IN1);
    float* outs = (float*)d_out;                        // single_out (B,16,32)
    float* outd = outs + (size_t)B * SH_B;              // double_out (B,16,16,8)

    const size_t g2bytes = (size_t)B * 128 * sizeof(float);
    const int use_ws = (ws_size >= 2 * g2bytes) ? 1 : 0;
    float* g2u = (float*)d_ws;
    float* g2d = g2u + (size_t)B * 128;

    fermi_double_kernel<<<B / (TB * MTA), 256, 0, stream>>>(dh, wm, cv, outd, g2u, g2d, use_ws);
    fermi_single_kernel<<<B / (TB * MTB), 256, 0, stream>>>(sh, dh, vm, bv, g2u, g2d, outs, use_ws);
}